// AttentionWithCommunity_44899588112465
// MI455X (gfx1250) — compile-verified
//
#include <hip/hip_runtime.h>

typedef __attribute__((ext_vector_type(2))) float v2f;
typedef __attribute__((ext_vector_type(8))) float v8f;

// Problem constants (match reference)
constexpr int N  = 20000;
constexpr int V  = 100000;
constexpr int C  = 5000;
constexpr int M  = 32;
constexpr int K  = 2048;
constexpr int D  = 128;
constexpr int F  = 3 * D;            // 384
constexpr int FS = F + 4;            // LDS feat stride (388) — avoids 64-bank conflicts
constexpr int HS = D + 4;            // hidden1 stride
constexpr int H2S = 64 + 4;          // hidden2 stride
constexpr int AW = (C + 31) / 32;    // active bitmap words (157)

__device__ __forceinline__ void fma4(float4& a, float w, const float4 e) {
    a.x += w * e.x; a.y += w * e.y; a.z += w * e.z; a.w += w * e.w;
}

__launch_bounds__(128, 1)
__global__ void attn_community_fused(
    const float* __restrict__ node_emb,      // [N,D]
    const float* __restrict__ member_score,  // [C,M]
    const float* __restrict__ cemb,          // [V,D]
    const float* __restrict__ W1,            // [3D,D]
    const float* __restrict__ b1,            // [D]
    const float* __restrict__ W2,            // [D]
    const float* __restrict__ b2,            // [1]
    const float* __restrict__ W3,            // [D,64]
    const float* __restrict__ b3,            // [64]
    const float* __restrict__ W4,            // [64]
    const float* __restrict__ b4,            // [1]
    const int*   __restrict__ node2community,// [V]
    const int*   __restrict__ community2node,// [C,M]
    const int*   __restrict__ member_num,    // [C]
    const int*   __restrict__ community_index,// [K]
    const int*   __restrict__ nodes,         // [N]
    float*       __restrict__ out)           // [N]
{
    __shared__ unsigned s_active[AW];
    __shared__ __align__(16) float s_feat[16 * FS];
    __shared__ float s_h1[16 * HS];
    __shared__ float s_h2[16 * H2S];
    __shared__ float s_w[16 * M];
    __shared__ int   s_mid[16 * M];
    __shared__ int   s_comm[16];
    __shared__ int   s_use[16];
    __shared__ int   s_node[16];

    const int tid    = threadIdx.x;       // 0..127
    const int block0 = blockIdx.x * 16;

    // ---- 1) build active-community bitmap in LDS ----
    for (int i = tid; i < AW; i += 128) s_active[i] = 0u;
    __syncthreads();
    for (int i = tid; i < K; i += 128) {
        int c = community_index[i];
        atomicOr(&s_active[c >> 5], 1u << (c & 31));
    }
    __syncthreads();

    // ---- 2) per-row metadata ----
    if (tid < 16) {
        int g = block0 + tid;
        int nd = 0, cm = 0, u = 0;
        if (g < N) {
            nd = nodes[g];
            cm = node2community[nd];
            u  = (s_active[cm >> 5] >> (cm & 31)) & 1;
        }
        s_node[tid] = nd;
        s_comm[tid] = cm;
        s_use[tid]  = (g < N) ? u : 0;
    }
    __syncthreads();

    // ---- 3) member weights + ids (512 entries, 4 per thread) ----
    for (int i = tid; i < 16 * M; i += 128) {
        int r = i >> 5, m = i & (M - 1);
        float w = 0.f; int mid = 0;
        if (s_use[r]) {
            int cm = s_comm[r];
            mid = community2node[cm * M + m];
            int nc = node2community[mid];
            bool valid = (m < member_num[cm]) &&
                         (((s_active[nc >> 5] >> (nc & 31)) & 1u) != 0u);
            if (valid) w = member_score[cm * M + m];
        }
        s_w[i]   = w;
        s_mid[i] = mid;
    }
    __syncthreads();

    // ---- 4) gather feat tile: 8 threads per row, 16 cols each ----
    {
        int r = tid >> 3, t = tid & 7;
        int col = t * 16;
        int g = block0 + r;
        float4 z4 = {0.f, 0.f, 0.f, 0.f};
        float4* fdst0 = (float4*)&s_feat[r * FS + col];           // node_emb part
        float4* fdst1 = (float4*)&s_feat[r * FS + D + col];       // community part
        float4* fdst2 = (float4*)&s_feat[r * FS + 2 * D + col];   // member part
        if (g < N) {
            const float4* s0 = (const float4*)(node_emb + (size_t)g * D + col);
            const float4* s1 = (const float4*)(cemb + (size_t)s_node[r] * D + col);
            #pragma unroll
            for (int i = 0; i < 4; ++i) { fdst0[i] = s0[i]; fdst1[i] = s1[i]; }
        } else {
            #pragma unroll
            for (int i = 0; i < 4; ++i) { fdst0[i] = z4; fdst1[i] = z4; }
        }
        float4 a0 = z4, a1 = z4, a2 = z4, a3 = z4;
        if (s_use[r]) {
            for (int m = 0; m < M; ++m) {
                float w = s_w[r * M + m];
                if (w != 0.f) {
                    const float4* e = (const float4*)(cemb + (size_t)s_mid[r * M + m] * D + col);
                    fma4(a0, w, e[0]); fma4(a1, w, e[1]);
                    fma4(a2, w, e[2]); fma4(a3, w, e[3]);
                }
            }
        }
        fdst2[0] = a0; fdst2[1] = a1; fdst2[2] = a2; fdst2[3] = a3;
    }
    __syncthreads();

    // ---- 5) GEMM1: hidden1 = relu(feat @ W1 + b1), via v_wmma_f32_16x16x4_f32 ----
    const int wave = tid >> 5;
    const int lane = tid & 31;
    const int hi   = lane >> 4;   // 0/1
    const int lm   = lane & 15;
    {
        v8f c0 = {}; v8f c1 = {};
        const int col0 = (2 * wave) * 16 + lm;
        const int col1 = col0 + 16;
        for (int k = 0; k < F; k += 4) {
            int kb = k + 2 * hi;
            v2f a;
            a.x = s_feat[lm * FS + kb];
            a.y = s_feat[lm * FS + kb + 1];
            v2f bA, bB;
            bA.x = W1[kb * D + col0];
            bA.y = W1[(kb + 1) * D + col0];
            bB.x = W1[kb * D + col1];
            bB.y = W1[(kb + 1) * D + col1];
            c0 = __builtin_amdgcn_wmma_f32_16x16x4_f32(false, a, false, bA,
                                                       (short)0, c0, false, false);
            c1 = __builtin_amdgcn_wmma_f32_16x16x4_f32(false, a, false, bB,
                                                       (short)0, c1, false, false);
        }
        float bb0 = b1[col0], bb1 = b1[col1];
        #pragma unroll
        for (int r8 = 0; r8 < 8; ++r8) {
            int row = r8 + 8 * hi;
            float h0 = c0[r8] + bb0;
            float h1 = c1[r8] + bb1;
            s_h1[row * HS + col0] = h0 > 0.f ? h0 : 0.f;
            s_h1[row * HS + col1] = h1 > 0.f ? h1 : 0.f;
        }
    }

    // ---- 5b) GEMM2: hidden2 = relu(node_emb @ W3 + b3) ----
    {
        v8f c0 = {};
        const int col = wave * 16 + lm;   // 0..63
        for (int k = 0; k < D; k += 4) {
            int kb = k + 2 * hi;
            v2f a;
            a.x = s_feat[lm * FS + kb];   // node_emb slice of feat
            a.y = s_feat[lm * FS + kb + 1];
            v2f bA;
            bA.x = W3[kb * 64 + col];
            bA.y = W3[(kb + 1) * 64 + col];
            c0 = __builtin_amdgcn_wmma_f32_16x16x4_f32(false, a, false, bA,
                                                       (short)0, c0, false, false);
        }
        float bb = b3[col];
        #pragma unroll
        for (int r8 = 0; r8 < 8; ++r8) {
            int row = r8 + 8 * hi;
            float h = c0[r8] + bb;
            s_h2[row * H2S + col] = h > 0.f ? h : 0.f;
        }
    }
    __syncthreads();

    // ---- 6) epilogue: dot with W2 / W4, select path ----
    if (tid < 16) {
        int g = block0 + tid;
        if (g < N) {
            float p1 = b2[0];
            for (int c = 0; c < D; ++c) p1 += s_h1[tid * HS + c] * W2[c];
            float p2 = b4[0];
            for (int c = 0; c < 64; ++c) p2 += s_h2[tid * H2S + c] * W4[c];
            out[g] = s_use[tid] ? p1 : p2;
        }
    }
}

extern "C" void kernel_launch(void* const* d_in, const int* in_sizes, int n_in,
                              void* d_out, int out_size, void* d_ws, size_t ws_size,
                              hipStream_t stream) {
    const float* node_emb        = (const float*)d_in[0];
    const float* member_score    = (const float*)d_in[1];
    const float* cemb            = (const float*)d_in[2];
    const float* W1              = (const float*)d_in[3];
    const float* b1              = (const float*)d_in[4];
    const float* W2              = (const float*)d_in[5];
    const float* b2              = (const float*)d_in[6];
    const float* W3              = (const float*)d_in[7];
    const float* b3              = (const float*)d_in[8];
    const float* W4              = (const float*)d_in[9];
    const float* b4              = (const float*)d_in[10];
    const int*   node2community  = (const int*)d_in[11];
    const int*   community2node  = (const int*)d_in[12];
    const int*   member_num      = (const int*)d_in[13];
    const int*   community_index = (const int*)d_in[14];
    const int*   nodes           = (const int*)d_in[15];
    float*       out             = (float*)d_out;

    int blocks = (N + 15) / 16;   // 1250
    attn_community_fused<<<blocks, 128, 0, stream>>>(
        node_emb, member_score, cemb, W1, b1, W2, b2, W3, b3, W4, b4,
        node2community, community2node, member_num, community_index, nodes, out);
}